// MiconiNetwork_80573586473189
// MI455X (gfx1250) — compile-verified
//
#include <hip/hip_runtime.h>
#include <math.h>

#define B    64
#define HS   1024
#define NIN  17
#define NACT 4

typedef __attribute__((ext_vector_type(2))) float        v2f;
typedef __attribute__((ext_vector_type(4))) float        v4f;
typedef __attribute__((ext_vector_type(8))) float        v8f;
typedef __attribute__((ext_vector_type(4))) unsigned int v4u;
typedef __attribute__((ext_vector_type(4))) int          v4i;
typedef __attribute__((ext_vector_type(8))) int          v8i;

// ---------------------------------------------------------------------------
// TDM: DMA one contiguous row of `nelem` f32 from global to LDS.
// D# built per CDNA5 ISA:
//   group0: count=1 | lds_addr[63:32] | global_addr[120:64] | type=2[127:126]
//   group1: data_size=4B | tensor_dim0=nelem | tensor_dim1=1 |
//           tile_dim0=nelem | tile_dim1=1 | tensor_dim0_stride=nelem
// This toolchain exposes the 6-arg builtin:
//   (uint32x4 g0, int32x8 g1, int32x4 g2, int32x4 g3, int32x8 g4, i32 cpol)
// Issued by one wave; completion via s_wait_tensorcnt then workgroup barrier.
// ---------------------------------------------------------------------------
#if __has_builtin(__builtin_amdgcn_tensor_load_to_lds)
#define USE_TDM 1
__device__ __forceinline__ void tdm_load_row_to_lds(const float* gsrc,
                                                    float* lds_dst, int nelem)
{
    unsigned long long ga = (unsigned long long)(uintptr_t)gsrc;
    unsigned int lds_addr = (unsigned int)(uintptr_t)lds_dst; // low32 = LDS byte addr

    v4u g0;
    g0.x = 1u;                                        // count=1, is_restore=0
    g0.y = lds_addr;                                  // bits 63:32
    g0.z = (unsigned int)(ga & 0xFFFFFFFFu);          // global_addr[31:0]
    g0.w = (unsigned int)((ga >> 32) & 0x01FFFFFFu)   // global_addr[56:32]
         | (2u << 30);                                // type=2 ("image")

    v8i g1;
    g1[0] = (2 << 16);                                // data_size: 2 -> 4 bytes
    g1[1] = (int)((unsigned)nelem << 16);             // tensor_dim0 low16 @ [63:48]
    g1[2] = (1 << 16);                                // dim0 hi16=0 | tensor_dim1=1
    g1[3] = (int)((unsigned)nelem << 16);             // dim1 hi16=0 | tile_dim0
    g1[4] = 1;                                        // tile_dim1=1, tile_dim2=0
    g1[5] = nelem;                                    // tensor_dim0_stride[31:0]
    g1[6] = 0;                                        // stride hi16 | dim1_stride lo
    g1[7] = 0;

    v4i z4 = {0, 0, 0, 0};
    v8i z8 = {0, 0, 0, 0, 0, 0, 0, 0};
    __builtin_amdgcn_tensor_load_to_lds(g0, g1, z4, z4, z8, 0);
}
#else
#define USE_TDM 0
#endif

// ---------------------------------------------------------------------------
// Kernel A: pre = inputs @ i2h_w.T + i2h_b   via V_WMMA_F32_16X16X4_F32
// M=64 (4 tiles), N=1024 (64 tiles), K=17 padded to 20 (5 wmma / tile).
// ---------------------------------------------------------------------------
__global__ void pre_wmma_kernel(const float* __restrict__ inputs,
                                const float* __restrict__ i2h_w,
                                const float* __restrict__ i2h_b,
                                float* __restrict__ pre)
{
    const int wave = threadIdx.x >> 5;
    const int lane = threadIdx.x & 31;
    const int tile = blockIdx.x * 8 + wave;     // 0..255
    const int mt = tile & 3;
    const int nt = tile >> 2;
    const int m0 = mt * 16;
    const int n0 = nt * 16;
    const int half = lane >> 4;
    const int lm   = lane & 15;

    v8f c = {};
    #pragma unroll
    for (int k0 = 0; k0 < 20; k0 += 4) {
        const int ka = k0 + 2 * half;
        v2f a, b;
        a.x = (ka     < NIN) ? inputs[(m0 + lm) * NIN + ka    ] : 0.0f;
        a.y = (ka + 1 < NIN) ? inputs[(m0 + lm) * NIN + ka + 1] : 0.0f;
        b.x = (ka     < NIN) ? i2h_w[(n0 + lm) * NIN + ka    ] : 0.0f;
        b.y = (ka + 1 < NIN) ? i2h_w[(n0 + lm) * NIN + ka + 1] : 0.0f;
        c = __builtin_amdgcn_wmma_f32_16x16x4_f32(false, a, false, b,
                                                  (short)0, c, false, false);
    }
    const float bias = i2h_b[n0 + lm];
    #pragma unroll
    for (int r = 0; r < 8; ++r) {
        const int m = m0 + r + half * 8;
        pre[m * HS + n0 + lm] = c[r] + bias;
    }
}

// ---------------------------------------------------------------------------
// Kernel B: fused single pass over hebb (512 MB total HBM traffic).
// Block = 8 waves sharing one row index i (different batches b):
//   * w[i,:], alpha[i,:] staged once into LDS via TDM (8 KB)
//   * each wave: hebb row (b,i,:) streamed NT into 32 VGPRs/lane,
//     plast reduction (shuffle-xor), tanh, NT writeback of hebb_new row.
// ---------------------------------------------------------------------------
__global__ void fused_hebb_kernel(const float* __restrict__ hebb,
                                  const float* __restrict__ hidden,
                                  const float* __restrict__ w,
                                  const float* __restrict__ alpha,
                                  const float* __restrict__ pre,
                                  const float* __restrict__ eta,
                                  float* __restrict__ hactiv_out,
                                  float* __restrict__ hebb_out)
{
    __shared__ __align__(16) float s_w[HS];
    __shared__ __align__(16) float s_a[HS];

    const int i    = blockIdx.x >> 3;           // hidden-unit row, shared by block
    const int bg   = blockIdx.x & 7;            // batch group
    const int wave = threadIdx.x >> 5;
    const int lane = threadIdx.x & 31;
    const int b    = bg * 8 + wave;             // this wave's batch
    const size_t row = (size_t)b * HS + i;

#if USE_TDM
    if (wave == 0) {                            // wave-uniform guard (TDM ignores EXEC)
        tdm_load_row_to_lds(w     + (size_t)i * HS, s_w, HS);
        tdm_load_row_to_lds(alpha + (size_t)i * HS, s_a, HS);
        __builtin_amdgcn_s_wait_tensorcnt(0);
    }
#else
    {
        const v4f* wsrc = (const v4f*)(w     + (size_t)i * HS);
        const v4f* asrc = (const v4f*)(alpha + (size_t)i * HS);
        for (int t = threadIdx.x; t < HS / 4; t += 256) {
            ((v4f*)s_w)[t] = wsrc[t];
            ((v4f*)s_a)[t] = asrc[t];
        }
    }
#endif
    __syncthreads();

    const v4f* hrow = (const v4f*)(hebb   + row * HS);
    const v4f* drow = (const v4f*)(hidden + (size_t)b * HS);
    const v4f* ws   = (const v4f*)s_w;
    const v4f* as   = (const v4f*)s_a;

    v4f hb[8];   // hebb row chunk kept in registers for writeback
    v4f hd[8];   // hidden chunk kept for writeback
    float acc = 0.0f;
    #pragma unroll
    for (int t = 0; t < 8; ++t) {
        const int j4 = t * 32 + lane;           // coalesced 16B/lane
        v4f hv = __builtin_nontemporal_load(hrow + j4);
        v4f wv = ws[j4];
        v4f av = as[j4];
        v4f dv = drow[j4];
        hb[t] = hv;
        hd[t] = dv;
        acc = fmaf(fmaf(av.x, hv.x, wv.x), dv.x, acc);
        acc = fmaf(fmaf(av.y, hv.y, wv.y), dv.y, acc);
        acc = fmaf(fmaf(av.z, hv.z, wv.z), dv.z, acc);
        acc = fmaf(fmaf(av.w, hv.w, wv.w), dv.w, acc);
    }

    // wave32 shuffle-xor reduction: all lanes end with the row sum
    #pragma unroll
    for (int off = 16; off > 0; off >>= 1)
        acc += __shfl_xor(acc, off, 32);

    const float h     = tanhf(pre[row] + acc);
    const float scale = eta[0] * h;

    v4f* orow = (v4f*)(hebb_out + row * HS);
    #pragma unroll
    for (int t = 0; t < 8; ++t) {
        const int j4 = t * 32 + lane;
        v4f o = hb[t] + scale * hd[t];
        __builtin_nontemporal_store(o, orow + j4);
    }
    if (lane == 0) hactiv_out[row] = h;
}

// ---------------------------------------------------------------------------
// Kernel C: heads as one [64x1024] x [1024x16] WMMA GEMM
// (cols 0-3 = h2o, col 4 = h2v). 4 blocks; 8 waves split K; LDS reduce.
// ---------------------------------------------------------------------------
__global__ void heads_wmma_kernel(const float* __restrict__ hactiv,
                                  const float* __restrict__ h2o_w,
                                  const float* __restrict__ h2o_b,
                                  const float* __restrict__ h2v_w,
                                  const float* __restrict__ h2v_b,
                                  float* __restrict__ activout,
                                  float* __restrict__ valueout)
{
    __shared__ float lds[8 * 256];
    const int wave = threadIdx.x >> 5;
    const int lane = threadIdx.x & 31;
    const int m0   = blockIdx.x * 16;
    const int half = lane >> 4;
    const int lm   = lane & 15;
    const int kbase = wave * 128;

    v8f c = {};
    for (int k0 = 0; k0 < 128; k0 += 4) {
        const int ka = kbase + k0 + 2 * half;
        v2f a, b;
        a.x = hactiv[(m0 + lm) * HS + ka];
        a.y = hactiv[(m0 + lm) * HS + ka + 1];
        float b0 = 0.0f, b1 = 0.0f;
        if (lm < NACT)       { b0 = h2o_w[lm * HS + ka]; b1 = h2o_w[lm * HS + ka + 1]; }
        else if (lm == NACT) { b0 = h2v_w[ka];           b1 = h2v_w[ka + 1]; }
        b.x = b0; b.y = b1;
        c = __builtin_amdgcn_wmma_f32_16x16x4_f32(false, a, false, b,
                                                  (short)0, c, false, false);
    }

    #pragma unroll
    for (int r = 0; r < 8; ++r)
        lds[wave * 256 + r * 32 + lane] = c[r];
    __syncthreads();

    const int t = threadIdx.x;
    float s = 0.0f;
    #pragma unroll
    for (int wv = 0; wv < 8; ++wv)
        s += lds[wv * 256 + t];
    const int r  = t >> 5;
    const int l5 = t & 31;
    const int hf = l5 >> 4;
    const int n  = l5 & 15;
    const int m  = m0 + r + hf * 8;
    if (n < NACT)       activout[m * NACT + n] = s + h2o_b[n];
    else if (n == NACT) valueout[m]            = s + h2v_b[0];
}

// ---------------------------------------------------------------------------
extern "C" void kernel_launch(void* const* d_in, const int* in_sizes, int n_in,
                              void* d_out, int out_size, void* d_ws, size_t ws_size,
                              hipStream_t stream)
{
    const float* inputs = (const float*)d_in[0];   // [B, NIN]
    const float* hidden = (const float*)d_in[1];   // [B, HS]
    const float* hebb   = (const float*)d_in[2];   // [B, HS, HS]
    // d_in[3] = et, d_in[4] = pw : unused by reference
    const float* i2h_w  = (const float*)d_in[5];   // [HS, NIN]
    const float* i2h_b  = (const float*)d_in[6];   // [HS]
    const float* w      = (const float*)d_in[7];   // [HS, HS]
    const float* alpha  = (const float*)d_in[8];   // [HS, HS]
    const float* eta    = (const float*)d_in[9];   // [1]
    const float* h2o_w  = (const float*)d_in[10];  // [NACT, HS]
    const float* h2o_b  = (const float*)d_in[11];  // [NACT]
    const float* h2v_w  = (const float*)d_in[12];  // [1, HS]
    const float* h2v_b  = (const float*)d_in[13];  // [1]

    float* out       = (float*)d_out;
    float* activout  = out;                         // B*NACT = 256
    float* valueout  = out + B * NACT;              // B      = 64
    float* hactiv    = out + B * NACT + B;          // B*HS   = 65536
    float* hebb_new  = out + B * NACT + B + B * HS; // B*HS*HS

    float* pre = (float*)d_ws;                      // B*HS floats (256 KB)

    pre_wmma_kernel<<<32, 256, 0, stream>>>(inputs, i2h_w, i2h_b, pre);

    fused_hebb_kernel<<<(B * HS) / 8, 256, 0, stream>>>(
        hebb, hidden, w, alpha, pre, eta, hactiv, hebb_new);

    heads_wmma_kernel<<<4, 256, 0, stream>>>(hactiv, h2o_w, h2o_b, h2v_w,
                                             h2v_b, activout, valueout);
}